// PrincipledBrdf_49503793054399
// MI455X (gfx1250) — compile-verified
//
#include <hip/hip_runtime.h>
#include <math.h>

#define PI_F      3.14159265358979323846f
#define INV_PI_F  0.31830988618379067154f
#define PTS       4      // points per thread
#define TPB       256    // 8 waves (wave32) per block

// Uniform, per-launch parameter block (computed once by a 1-thread setup
// kernel into d_ws, read via scalar loads by the main kernel).
struct BrdfParams {
    float cd_lin0, cd_lin1, cd_lin2;        // base_color^2.2
    float cs0, cs1, cs2;                    // c_spec0
    float sh0, sh1, sh2;                    // sheen * c_sheen
    float one_minus_mtl;
    float sub;
    float rough;
    float inv_ax2, inv_ay2;                 // 1/ax^2, 1/ay^2
    float inv_pi_axay;                      // 1/(pi*ax*ay)
    float ax2, ay2;                         // ax^2, ay^2
    float dr_a2m1;                          // (a2-1), or 0 when a_cc>=1
    float cc_dr;                            // 0.25*cc * Dr_numerator
};

__global__ void brdf_setup_kernel(const float* __restrict__ base_color,
                                  const float* __restrict__ metallic,
                                  const float* __restrict__ subsurface,
                                  const float* __restrict__ specular,
                                  const float* __restrict__ roughness,
                                  const float* __restrict__ specular_tint,
                                  const float* __restrict__ anisotropic,
                                  const float* __restrict__ sheen,
                                  const float* __restrict__ sheen_tint,
                                  const float* __restrict__ clear_coat,
                                  const float* __restrict__ clear_coat_gloss,
                                  BrdfParams* __restrict__ P)
{
    if (threadIdx.x != 0 || blockIdx.x != 0) return;

    const float mtl     = metallic[0];
    const float sub     = subsurface[0];
    const float spec    = specular[0];
    const float rough   = roughness[0];
    const float sp_tint = specular_tint[0];
    const float aniso   = anisotropic[0];
    const float sh      = sheen[0];
    const float sh_tint = sheen_tint[0];
    const float cc      = clear_coat[0];
    const float ccg     = clear_coat_gloss[0];

    float cd[3];
    cd[0] = powf(base_color[0], 2.2f);
    cd[1] = powf(base_color[1], 2.2f);
    cd[2] = powf(base_color[2], 2.2f);
    const float lum = 0.3f * cd[0] + 0.6f * cd[1] + 0.1f * cd[2];
    const float inv_lum = 1.0f / fmaxf(lum, 1e-20f);
    float ct[3];
    for (int i = 0; i < 3; ++i) ct[i] = (lum > 0.0f) ? cd[i] * inv_lum : 0.0f;

    float cs[3], shc[3];
    for (int i = 0; i < 3; ++i) {
        const float tinted = 1.0f + sp_tint * (ct[i] - 1.0f);     // lerp(1, c_tint, sp_tint)
        const float s0 = spec * 0.08f * tinted;
        cs[i]  = s0 + mtl * (cd[i] - s0);                          // lerp(s0, cd_lin, mtl)
        shc[i] = sh * (1.0f + sh_tint * (ct[i] - 1.0f));           // sheen * lerp(1, c_tint, sh_tint)
    }

    const float aspect = sqrtf(1.0f - aniso * 0.9f);
    const float r2 = rough * rough;
    const float ax = fmaxf(0.001f, r2 / aspect);
    const float ay = fmaxf(0.001f, r2 * aspect);

    const float a_cc = 0.1f + (0.001f - 0.1f) * ccg;               // lerp(0.1, 0.001, ccg)
    const float a2 = a_cc * a_cc;
    float dr_num, dr_a2m1;
    if (a_cc >= 1.0f) { dr_num = INV_PI_F; dr_a2m1 = 0.0f; }
    else              { dr_num = (a2 - 1.0f) / (PI_F * logf(a2)); dr_a2m1 = a2 - 1.0f; }

    P->cd_lin0 = cd[0];  P->cd_lin1 = cd[1];  P->cd_lin2 = cd[2];
    P->cs0 = cs[0];      P->cs1 = cs[1];      P->cs2 = cs[2];
    P->sh0 = shc[0];     P->sh1 = shc[1];     P->sh2 = shc[2];
    P->one_minus_mtl = 1.0f - mtl;
    P->sub = sub;
    P->rough = rough;
    P->inv_ax2 = 1.0f / (ax * ax);
    P->inv_ay2 = 1.0f / (ay * ay);
    P->inv_pi_axay = 1.0f / (PI_F * ax * ay);
    P->ax2 = ax * ax;
    P->ay2 = ay * ay;
    P->dr_a2m1 = dr_a2m1;
    P->cc_dr = 0.25f * cc * dr_num;
}

// clamp01 in a single VALU op (v_med3_f32)
__device__ __forceinline__ float clamp01(float x) {
    return __builtin_amdgcn_fmed3f(x, 0.0f, 1.0f);
}
__device__ __forceinline__ float pow5(float m) {
    const float m2 = m * m;
    return m2 * m2 * m;
}

// Derived per-thread constants (SALU, once per thread)
struct BrdfDerived {
    float kdc;            // (1/pi) * (1 - mtl)
    float ics0, ics1, ics2; // 1 - c_spec0
};

__device__ __forceinline__ void brdf_point(const BrdfParams& P, const BrdfDerived& D,
                                           float lx, float ly, float lz,
                                           float vx, float vy, float vz,
                                           float& r0, float& r1, float& r2)
{
    // Half vector
    float hx = lx + vx, hy = ly + vy, hz = lz + vz;
    const float inv_hn = __builtin_amdgcn_rsqf(hx * hx + hy * hy + hz * hz);
    hx *= inv_hn; hy *= inv_hn; hz *= inv_hn;

    const float chl = hx * lx + hy * ly + hz * lz;   // cos(half, light)
    const bool valid = (lz >= 0.0f) && (vz >= 0.0f);
    const float cnl = valid ? lz : 0.5f;
    const float cnv = valid ? vz : 0.5f;

    const float fl = pow5(clamp01(1.0f - cnl));
    const float fv = pow5(clamp01(1.0f - cnv));
    const float FH = pow5(clamp01(1.0f - chl));

    // Diffuse retro-reflection + subsurface
    const float chl2r  = chl * chl * P.rough;
    const float fd90m1 = 2.0f * chl2r - 0.5f;                        // fd90 - 1
    const float fd  = (1.0f + fd90m1 * fl) * (1.0f + fd90m1 * fv);
    const float fssm1 = chl2r - 1.0f;                                // fss90 - 1
    const float fss = (1.0f + fssm1 * fl) * (1.0f + fssm1 * fv);
    const float ss  = 1.25f * (fss * (__builtin_amdgcn_rcpf(cnl + cnv) - 0.5f) + 0.5f);

    // Anisotropic GGX specular: Gs*Ds with a single fused reciprocal
    const float Dden = hx * hx * P.inv_ax2 + hy * hy * P.inv_ay2 + hz * hz;
    const float gl = cnl + __builtin_amdgcn_sqrtf(lx * lx * P.ax2 + ly * ly * P.ay2 + cnl * cnl);
    const float gv = cnv + __builtin_amdgcn_sqrtf(vx * vx * P.ax2 + vy * vy * P.ay2 + cnv * cnv);
    const float GsDs = P.inv_pi_axay * __builtin_amdgcn_rcpf(Dden * Dden * gl * gv);

    // Clear coat: 0.25*cc*Gr*Fr*Dr with a single fused reciprocal
    const float tcc = 1.0f + P.dr_a2m1 * hz * hz;
    const float grl = cnl + __builtin_amdgcn_sqrtf(0.0625f + cnl * cnl * 0.9375f);
    const float grv = cnv + __builtin_amdgcn_sqrtf(0.0625f + cnv * cnv * 0.9375f);
    const float Fr  = 0.04f + 0.96f * FH;
    const float clear = P.cc_dr * Fr * __builtin_amdgcn_rcpf(tcc * grl * grv);

    const float mixd   = fd + (ss - fd) * P.sub;                     // lerp(fd, ss, sub)
    const float kd     = mixd * D.kdc;
    const float ksheen = FH * P.one_minus_mtl;

    const float o0 = kd * P.cd_lin0 + ksheen * P.sh0
                   + GsDs * (P.cs0 + FH * D.ics0) + clear;
    const float o1 = kd * P.cd_lin1 + ksheen * P.sh1
                   + GsDs * (P.cs1 + FH * D.ics1) + clear;
    const float o2 = kd * P.cd_lin2 + ksheen * P.sh2
                   + GsDs * (P.cs2 + FH * D.ics2) + clear;

    r0 = valid ? o0 : 0.0f;
    r1 = valid ? o1 : 0.0f;
    r2 = valid ? o2 : 0.0f;
}

__global__ void __launch_bounds__(TPB)
brdf_main_kernel(const float4* __restrict__ in4,
                 const float*  __restrict__ inf,
                 float4*       __restrict__ out4,
                 float*        __restrict__ outf,
                 const BrdfParams* __restrict__ Pp,
                 int npts)
{
    const BrdfParams P = *Pp;   // uniform address -> scalar loads

    BrdfDerived D;
    D.kdc  = INV_PI_F * P.one_minus_mtl;
    D.ics0 = 1.0f - P.cs0;
    D.ics1 = 1.0f - P.cs1;
    D.ics2 = 1.0f - P.cs2;

    const int t = blockIdx.x * blockDim.x + threadIdx.x;
    const long base = (long)t * PTS;
    if (base >= npts) return;

    if (base + PTS <= npts) {
        // Fast path: 4 points = 24 floats in (6x b128), 12 floats out (3x b128)
        const long fi = (base * 6) >> 2;   // base % 4 == 0 on the fast-path grid
        const float4 q0 = in4[fi + 0];
        const float4 q1 = in4[fi + 1];
        const float4 q2 = in4[fi + 2];
        const float4 q3 = in4[fi + 3];
        const float4 q4 = in4[fi + 4];
        const float4 q5 = in4[fi + 5];
        const float f[24] = { q0.x, q0.y, q0.z, q0.w,
                              q1.x, q1.y, q1.z, q1.w,
                              q2.x, q2.y, q2.z, q2.w,
                              q3.x, q3.y, q3.z, q3.w,
                              q4.x, q4.y, q4.z, q4.w,
                              q5.x, q5.y, q5.z, q5.w };
        float o[12];
#pragma unroll
        for (int j = 0; j < PTS; ++j) {
            brdf_point(P, D,
                       f[6 * j + 0], f[6 * j + 1], f[6 * j + 2],
                       f[6 * j + 3], f[6 * j + 4], f[6 * j + 5],
                       o[3 * j + 0], o[3 * j + 1], o[3 * j + 2]);
        }
        const long oi = (base * 3) >> 2;
        out4[oi + 0] = make_float4(o[0], o[1], o[2],  o[3]);
        out4[oi + 1] = make_float4(o[4], o[5], o[6],  o[7]);
        out4[oi + 2] = make_float4(o[8], o[9], o[10], o[11]);
    } else {
        // Tail (not taken for N = 4194304, kept for generality)
        for (long p = base; p < npts; ++p) {
            float r0, r1, r2;
            brdf_point(P, D,
                       inf[6 * p + 0], inf[6 * p + 1], inf[6 * p + 2],
                       inf[6 * p + 3], inf[6 * p + 4], inf[6 * p + 5],
                       r0, r1, r2);
            outf[3 * p + 0] = r0;
            outf[3 * p + 1] = r1;
            outf[3 * p + 2] = r2;
        }
    }
}

extern "C" void kernel_launch(void* const* d_in, const int* in_sizes, int n_in,
                              void* d_out, int out_size, void* d_ws, size_t ws_size,
                              hipStream_t stream)
{
    const float* inputs = (const float*)d_in[0];
    BrdfParams*  P      = (BrdfParams*)d_ws;

    brdf_setup_kernel<<<1, 1, 0, stream>>>(
        (const float*)d_in[1],  // base_color
        (const float*)d_in[2],  // metallic
        (const float*)d_in[3],  // subsurface
        (const float*)d_in[4],  // specular
        (const float*)d_in[5],  // roughness
        (const float*)d_in[6],  // specular_tint
        (const float*)d_in[7],  // anisotropic
        (const float*)d_in[8],  // sheen
        (const float*)d_in[9],  // sheen_tint
        (const float*)d_in[10], // clear_coat
        (const float*)d_in[11], // clear_coat_gloss
        P);

    const int npts = in_sizes[0] / 6;   // inputs: [N, 2, 3] f32
    const long threads = ((long)npts + PTS - 1) / PTS;
    const int blocks = (int)((threads + TPB - 1) / TPB);

    brdf_main_kernel<<<blocks, TPB, 0, stream>>>(
        (const float4*)inputs, inputs,
        (float4*)d_out, (float*)d_out,
        P, npts);
}